// BSplineBasis_70446053589597
// MI455X (gfx1250) — compile-verified
//
#include <hip/hip_runtime.h>

#ifndef __has_builtin
#define __has_builtin(x) 0
#endif

#if __has_builtin(__builtin_amdgcn_global_load_async_to_lds_b32)
#define ASYNC_LDS 1
#else
#define ASYNC_LDS 0
#endif

#define THREADS 256
#define VEC     4                  // splines per thread (float4 stream)
#define STILE   (THREADS * VEC)    // 1024 splines per block
#define BTILE   32                 // batch rows per block (coef reuse = 32x)
#define KCOEF   8                  // GRID_SIZE + SPLINE_ORDER
#define CSTRIDE 9                  // LDS row stride (floats): 36t mod 64 has period 16
                                   // -> at most 2-way bank conflict on the 4-wide gather

#if ASYNC_LDS
typedef __attribute__((address_space(1))) int GI;   // global (AS1)
typedef __attribute__((address_space(3))) int LI;   // LDS (AS3)
#endif

struct SplineParams { float g0, h, i1, i2, i3, invh; };

__device__ __forceinline__ float eval_one(float x, const float* __restrict__ cl,
                                          const SplineParams p) {
    // Knot interval: grid[j] <= x < grid[j+1]; cubic support window = [j-3, j].
    int j = (int)((x - p.g0) * p.invh);
    j = (j < 3) ? 3 : j;
    j = (j > KCOEF - 1) ? (KCOEF - 1) : j;
    const float h = p.h;
    float a = x - fmaf((float)j, h, p.g0);          // a in [0, h)
    // Windowed Cox-de Boor over the 4 non-zero bases (uniform knots):
    float b1v = a * p.i1;
    float b0v = (h - a) * p.i1;
    float c0 = (h - a) * p.i2 * b0v;
    float c1 = ((a + h) * b0v + (2.0f * h - a) * b1v) * p.i2;
    float c2 = a * p.i2 * b1v;
    float n0 = (h - a) * p.i3 * c0;
    float n1 = ((a + 2.0f * h) * c0 + (2.0f * h - a) * c1) * p.i3;
    float n2 = ((a + h) * c1 + (3.0f * h - a) * c2) * p.i3;
    float n3 = a * p.i3 * c2;
    const float* w = cl + (j - 3);                  // per-lane LDS gather
    return n0 * w[0] + n1 * w[1] + n2 * w[2] + n3 * w[3];
}

__launch_bounds__(THREADS, 4)
__global__ void bspline_eval_kernel(const float* __restrict__ X,
                                    const float* __restrict__ C,
                                    const float* __restrict__ G,
                                    float* __restrict__ Out,
                                    int B, int S) {
    __shared__ float lds[STILE * CSTRIDE];          // 36 KB coefficient tile

    const int t  = threadIdx.x;
    const int s0 = blockIdx.x * STILE;
    const int b0 = blockIdx.y * BTILE;

    // ---- Stage coefficient rows for this s-tile into LDS (async DMA) ----
    #pragma unroll
    for (int q = 0; q < VEC; ++q) {
        const int r = t + THREADS * q;              // lane-dense row index
        if (s0 + r < S) {
            const float* gsrc = C + (size_t)(s0 + r) * KCOEF;
            float*       ldst = &lds[r * CSTRIDE];
#if ASYNC_LDS
            GI* g = (GI*)gsrc;
            LI* l = (LI*)ldst;
            // IOFFSET (immediate) is added to both global and LDS addresses.
            __builtin_amdgcn_global_load_async_to_lds_b32(g, l, 0, 0);
            __builtin_amdgcn_global_load_async_to_lds_b32(g, l, 4, 0);
            __builtin_amdgcn_global_load_async_to_lds_b32(g, l, 8, 0);
            __builtin_amdgcn_global_load_async_to_lds_b32(g, l, 12, 0);
            __builtin_amdgcn_global_load_async_to_lds_b32(g, l, 16, 0);
            __builtin_amdgcn_global_load_async_to_lds_b32(g, l, 20, 0);
            __builtin_amdgcn_global_load_async_to_lds_b32(g, l, 24, 0);
            __builtin_amdgcn_global_load_async_to_lds_b32(g, l, 28, 0);
#else
            #pragma unroll
            for (int k = 0; k < KCOEF; ++k) ldst[k] = gsrc[k];
#endif
        }
    }
#if ASYNC_LDS
#if __has_builtin(__builtin_amdgcn_s_wait_asynccnt)
    __builtin_amdgcn_s_wait_asynccnt(0);
#else
    asm volatile("s_wait_asynccnt 0" ::: "memory");
#endif
#endif
    __syncthreads();

    // ---- Uniform-knot parameters (grid is linspace, h constant) ----
    SplineParams p;
    p.g0 = G[0];
    p.h  = G[1] - p.g0;
    const float eps = 1e-8f;
    p.i1 = 1.0f / (p.h + eps);                      // reference: k*h + eps denominators
    p.i2 = 1.0f / (2.0f * p.h + eps);
    p.i3 = 1.0f / (3.0f * p.h + eps);
    p.invh = 1.0f / p.h;

    const int sbase = s0 + t * VEC;
    const float* cl0 = &lds[(t * VEC) * CSTRIDE];

    if ((S % VEC == 0) && (sbase + VEC <= S) && (b0 + BTILE <= B)) {
        // ---- Fast path: b128 streaming, unrollable (no breaks) ----
        const int strideV = S / VEC;
        const float4* xp = (const float4*)(X   + (size_t)b0 * S + sbase);
        float4*       op = (float4*)      (Out + (size_t)b0 * S + sbase);
        #pragma unroll 8
        for (int i = 0; i < BTILE; ++i) {
            float4 xv = xp[(size_t)i * strideV];
            if (i + 8 < BTILE)
                __builtin_prefetch(&xp[(size_t)(i + 8) * strideV], 0, 0);
            float4 r;
            r.x = eval_one(xv.x, cl0 + 0 * CSTRIDE, p);
            r.y = eval_one(xv.y, cl0 + 1 * CSTRIDE, p);
            r.z = eval_one(xv.z, cl0 + 2 * CSTRIDE, p);
            r.w = eval_one(xv.w, cl0 + 3 * CSTRIDE, p);
            op[(size_t)i * strideV] = r;
        }
    } else {
        // ---- Remainder path: scalar, bounds-checked ----
        for (int i = 0; i < BTILE; ++i) {
            if (b0 + i >= B) break;
            for (int q = 0; q < VEC; ++q) {
                const int s = sbase + q;
                if (s < S) {
                    float x = X[(size_t)(b0 + i) * S + s];
                    Out[(size_t)(b0 + i) * S + s] =
                        eval_one(x, cl0 + q * CSTRIDE, p);
                }
            }
        }
    }
}

extern "C" void kernel_launch(void* const* d_in, const int* in_sizes, int n_in,
                              void* d_out, int out_size, void* d_ws, size_t ws_size,
                              hipStream_t stream) {
    const float* x    = (const float*)d_in[0];   // (B, S)
    const float* coef = (const float*)d_in[1];   // (S, 8)
    const float* grid = (const float*)d_in[2];   // (12,)
    float* out = (float*)d_out;                  // (B, S)

    const int S = in_sizes[1] / KCOEF;
    const int B = in_sizes[0] / S;

    dim3 block(THREADS);
    dim3 grd((S + STILE - 1) / STILE, (B + BTILE - 1) / BTILE);
    bspline_eval_kernel<<<grd, block, 0, stream>>>(x, coef, grid, out, B, S);
}